// SpatialDynamicExtractor_6975026888820
// MI455X (gfx1250) — compile-verified
//
#include <hip/hip_runtime.h>

typedef __attribute__((ext_vector_type(2))) float v2f;
typedef __attribute__((ext_vector_type(8))) float v8f;

#define LDS_LD 68   // 64 + 4 pad: fragment row-reads advance 4 banks/row -> conflict-free

__global__ __launch_bounds__(256)
void sde_scores_kernel(const float* __restrict__ x_in,   // (B,T,N,D) = (8,512,64,64)
                       const float* __restrict__ pos,    // (N,D) = (64,64)
                       const float* __restrict__ gamma,  // (64)
                       const float* __restrict__ beta,   // (64)
                       const float* __restrict__ Wq,     // (64,64) row-major (e,d)
                       const float* __restrict__ Wk,     // (64,64)
                       float* __restrict__ out)          // (B,T,H,N,N) = (8,512,8,64,64)
{
    __shared__ float x_s[64 * LDS_LD];
    __shared__ float q_s[64 * LDS_LD];
    __shared__ float k_s[64 * LDS_LD];

    const int tid  = threadIdx.x;
    const int bt   = blockIdx.x;               // fused (b,t), 0..4095
    const size_t in_base = (size_t)bt * (64 * 64);

    // ---------------- Stage 1: add pos_embed + LayerNorm -> x_s ----------------
    {
        const int row = tid >> 2;              // 0..63
        const int seg = (tid & 3) * 16;        // 16 floats per lane
        const float* src = x_in + in_base + row * 64 + seg;
        const float* pe  = pos + row * 64 + seg;

        float vals[16];
        float s = 0.f, ss = 0.f;
        #pragma unroll
        for (int i = 0; i < 4; ++i) {
            float4 a = reinterpret_cast<const float4*>(src)[i];
            float4 p = reinterpret_cast<const float4*>(pe)[i];
            a.x += p.x; a.y += p.y; a.z += p.z; a.w += p.w;
            vals[i*4+0] = a.x; vals[i*4+1] = a.y; vals[i*4+2] = a.z; vals[i*4+3] = a.w;
            s  += a.x + a.y + a.z + a.w;
            ss += a.x*a.x + a.y*a.y + a.z*a.z + a.w*a.w;
        }
        // reduce across the 4 lanes owning this row (lanes differ only in bits 0..1)
        s += __shfl_xor(s, 1);  ss += __shfl_xor(ss, 1);
        s += __shfl_xor(s, 2);  ss += __shfl_xor(ss, 2);

        const float mu   = s * (1.f / 64.f);
        const float var  = ss * (1.f / 64.f) - mu * mu;
        const float rstd = rsqrtf(var + 1e-5f);
        #pragma unroll
        for (int i = 0; i < 16; ++i) {
            const int d = seg + i;
            x_s[row * LDS_LD + d] = (vals[i] - mu) * rstd * gamma[d] + beta[d];
        }
    }
    __syncthreads();

    const int wave = tid >> 5;                 // 0..7
    const int lane = tid & 31;
    const int l    = lane & 15;                // fragment row/col within half-wave
    const int hi   = lane >> 4;                // 0 or 1
    const int koff = hi * 2;                   // K sub-offset for A/B fragments

    // ---------------- Stage 2: Q = x @ Wq^T, K = x @ Wk^T (WMMA f32 16x16x4) ----------------
    // 32 output tiles of 16x16 (16 for Q, 16 for K); 4 tiles per wave.
    #pragma unroll
    for (int i = 0; i < 4; ++i) {
        const int t   = wave * 4 + i;          // 0..31
        const int isK = t >> 4;
        const int tt  = t & 15;
        const int m0  = (tt >> 2) * 16;        // token-row tile
        const int e0  = (tt & 3) * 16;         // output-feature tile
        const float* W = isK ? Wk : Wq;
        float* dst     = isK ? k_s : q_s;

        v8f acc = {};
        #pragma unroll
        for (int k0 = 0; k0 < 64; k0 += 4) {
            v2f a, b;
            const float* xp = &x_s[(m0 + l) * LDS_LD + k0 + koff];
            a.x = xp[0]; a.y = xp[1];
            const float* wp = W + (size_t)(e0 + l) * 64 + k0 + koff;   // B[k][e] = W[e][k]
            b.x = wp[0]; b.y = wp[1];
            acc = __builtin_amdgcn_wmma_f32_16x16x4_f32(false, a, false, b,
                                                        (short)0, acc, false, false);
        }
        #pragma unroll
        for (int r = 0; r < 8; ++r)
            dst[(m0 + r + hi * 8) * LDS_LD + e0 + l] = acc[r];
    }
    __syncthreads();

    // ---------------- Stage 3: per-head scores = Q_h @ K_h^T, scaled ----------------
    const float scale = 0.35355339059327373f;  // 1/sqrt(8)
    const int h = wave;                        // one head per wave
    float* outh = out + ((size_t)bt * 8 + h) * 4096;

    #pragma unroll
    for (int j = 0; j < 16; ++j) {
        const int n0 = (j >> 2) * 16;
        const int m0 = (j & 3) * 16;
        v8f acc = {};
        #pragma unroll
        for (int kk = 0; kk < 2; ++kk) {
            const int k0 = h * 8 + kk * 4;
            v2f a, b;
            const float* qp = &q_s[(n0 + l) * LDS_LD + k0 + koff];
            a.x = qp[0]; a.y = qp[1];
            const float* kp = &k_s[(m0 + l) * LDS_LD + k0 + koff];   // B[d][m] = K[m][h*8+d]
            b.x = kp[0]; b.y = kp[1];
            acc = __builtin_amdgcn_wmma_f32_16x16x4_f32(false, a, false, b,
                                                        (short)0, acc, false, false);
        }
        #pragma unroll
        for (int r = 0; r < 8; ++r) {
            const int n = n0 + r + hi * 8;
            outh[(size_t)n * 64 + m0 + l] = acc[r] * scale;
        }
    }
}

extern "C" void kernel_launch(void* const* d_in, const int* in_sizes, int n_in,
                              void* d_out, int out_size, void* d_ws, size_t ws_size,
                              hipStream_t stream) {
    const float* x     = (const float*)d_in[0];
    const float* pos   = (const float*)d_in[1];
    const float* gamma = (const float*)d_in[2];
    const float* beta  = (const float*)d_in[3];
    const float* Wq    = (const float*)d_in[4];
    const float* Wk    = (const float*)d_in[5];
    float* out = (float*)d_out;

    dim3 grid(8 * 512);   // one block per (b,t)
    dim3 block(256);      // 8 waves (wave32)
    hipLaunchKernelGGL(sde_scores_kernel, grid, block, 0, stream,
                       x, pos, gamma, beta, Wq, Wk, out);
}